// RayCastLayer_5463198400791
// MI455X (gfx1250) — compile-verified
//
#include <hip/hip_runtime.h>
#include <hip/hip_bf16.h>

// RayCastLayer on MI455X (gfx1250):
// out[b,c,p] = sum_q W[p,q] * x[b,c,q]  ->  GEMM (2048 x 361) * (361 x 361)^T
// using V_WMMA_F32_16X16X4_F32 with the dense ray-weight matrix built on device.

typedef float v2f __attribute__((ext_vector_type(2)));
typedef float v8f __attribute__((ext_vector_type(8)));

#define BOARD 19
#define NPOS  361          // 19*19
#define MROWS 2048         // 32*64 (b,c) rows
#define PAD   368          // 23*16, covers both N-pad (361->368) and K-pad
#define NTILES 23          // ceil(361/16)
#define MTILES 128         // 2048/16

// ---------------------------------------------------------------------------
// Kernel 1: build Bt[n][k] = W[p=n, q=k], zero-padded to PAD x PAD.
// W[p,q] = weight[0][d-1] if q-p is an axial offset of distance d,
//          weight[1][d-1] if diagonal, else 0. In-bounds is implied since
//          both p and q index the board. One thread per element; no atomics.
// ---------------------------------------------------------------------------
__global__ __launch_bounds__(256) void raycast_build_bt(
    const float* __restrict__ weight,   // [2][18]
    float* __restrict__ Bt)             // [PAD][PAD], row = n (out pos), col = k (src pos)
{
    const int t = blockIdx.x * blockDim.x + threadIdx.x;   // 0 .. PAD*PAD-1
    const int n = t / PAD;  // output position p
    const int k = t % PAD;  // source position q
    float w = 0.0f;
    if (n < NPOS && k < NPOS) {
        const int py = n / BOARD, px = n % BOARD;
        const int qy = k / BOARD, qx = k % BOARD;
        const int dy = qy - py,  dx = qx - px;
        const int ay = dy < 0 ? -dy : dy;
        const int ax = dx < 0 ? -dx : dx;
        if (dy == 0 && dx != 0) {
            w = weight[0 * 18 + (ax - 1)];          // horizontal line ray
        } else if (dx == 0 && dy != 0) {
            w = weight[0 * 18 + (ay - 1)];          // vertical line ray
        } else if (ax == ay && ax != 0) {
            w = weight[1 * 18 + (ax - 1)];          // diagonal ray
        }
    }
    Bt[t] = w;
}

// ---------------------------------------------------------------------------
// Kernel 2: GEMM via V_WMMA_F32_16X16X4_F32. One wave32 per 16x16 out tile.
//   A frag (16x4 f32, 2 VGPR): lane m = lane&15, K-pair = lane>>4,
//                              vgpr j holds K = k0 + 2*(lane>>4) + j
//   B frag (4x16 f32, 2 VGPR): lane n = lane&15, same K-pair split
//   C/D   (16x16 f32, 8 VGPR): vgpr r -> M = r + 8*(lane>>4), N = lane&15
// ---------------------------------------------------------------------------
__global__ __launch_bounds__(256) void raycast_gemm_wmma(
    const float* __restrict__ X,    // [MROWS][NPOS] row-major (b,c flattened)
    const float* __restrict__ Bt,   // [PAD][PAD]   row = n, col = k
    float* __restrict__ Out)        // [MROWS][NPOS]
{
    const int tid   = threadIdx.x;
    const int lane  = tid & 31;
    const int gwave = blockIdx.x * 8 + (tid >> 5);   // 0 .. 2943
    const int mtile = gwave / NTILES;                // 0 .. 127
    const int ntile = gwave % NTILES;                // 0 .. 22

    const int l16   = lane & 15;
    const int khalf = lane >> 4;                     // 0 or 1

    // Per-lane row pointers (A uses l16 as M row; B uses l16 as N column-row of Bt)
    const float* __restrict__ xr = X  + (mtile * 16 + l16) * NPOS;   // 4B aligned
    const float* __restrict__ br = Bt + (ntile * 16 + l16) * PAD;    // 16B aligned

    v8f acc = {};

    // Main K loop: k0 = 0..356 step 4  (k up to 359 < 361, no guards needed)
    #pragma unroll 6
    for (int k0 = 0; k0 < 360; k0 += 4) {
        const int ka = k0 + 2 * khalf;
        v2f a;
        a.x = xr[ka];
        a.y = xr[ka + 1];
        const v2f b = *(const v2f*)(br + ka);        // ka even -> 8B aligned
        acc = __builtin_amdgcn_wmma_f32_16x16x4_f32(
            /*neg_a=*/false, a, /*neg_b=*/false, b,
            /*c_mod=*/(short)0, acc, /*reuse_a=*/false, /*reuse_b=*/false);
    }

    // Tail: k0 = 360 covers k = 360..363; only k==360 exists in X.
    // Bt is zero-padded for k >= 361, so masked-zero A lanes contribute 0.
    {
        const int ka = 360 + 2 * khalf;              // 360 or 362
        v2f a;
        a.x = (ka < NPOS) ? xr[ka] : 0.0f;
        a.y = 0.0f;                                  // ka+1 >= 361 always
        const v2f b = *(const v2f*)(br + ka);
        acc = __builtin_amdgcn_wmma_f32_16x16x4_f32(
            false, a, false, b, (short)0, acc, false, false);
    }

    // Store D: lane l16 = N column, vgpr r -> M = r + 8*khalf
    const int n = ntile * 16 + l16;
    if (n < NPOS) {
        const int mbase = mtile * 16 + khalf * 8;
        #pragma unroll
        for (int r = 0; r < 8; ++r) {
            Out[(mbase + r) * NPOS + n] = acc[r];
        }
    }
}

// ---------------------------------------------------------------------------
// Launch: inputs are {x, weight, flat_index, valid_mask}; geometry is static
// so we rebuild W analytically from `weight` alone (matches reference exactly).
// ---------------------------------------------------------------------------
extern "C" void kernel_launch(void* const* d_in, const int* in_sizes, int n_in,
                              void* d_out, int out_size, void* d_ws, size_t ws_size,
                              hipStream_t stream) {
    const float* x      = (const float*)d_in[0];   // [32,64,19,19] = [2048][361]
    const float* weight = (const float*)d_in[1];   // [2][18]
    float*       out    = (float*)d_out;           // [2048][361]
    float*       Bt     = (float*)d_ws;            // PAD*PAD floats = 529 KB

    // Build padded weight matrix: PAD*PAD = 135424 threads = 529 blocks x 256
    raycast_build_bt<<<(PAD * PAD) / 256, 256, 0, stream>>>(weight, Bt);

    // GEMM: 2944 wave-tiles / 8 waves per block = 368 blocks
    raycast_gemm_wmma<<<(MTILES * NTILES) / 8, 256, 0, stream>>>(x, Bt, out);
}